// NodeAttentionLayer_71536975282976
// MI455X (gfx1250) — compile-verified
//
#include <hip/hip_runtime.h>
#include <hip/hip_bf16.h>
#include <math.h>

typedef float v2f __attribute__((ext_vector_type(2)));
typedef float v8f __attribute__((ext_vector_type(8)));

#define NEG_SLOPE 0.2f
#define HEADS 8
#define DIMD 32
#define HD 256     // HEADS*DIMD
#define DIN 128

// ---------------------------------------------------------------------------
// fp32 WMMA GEMM: ft[N, 256] = feats[N, 128] @ W[128, 256]
// One wave computes a 64x64 output tile: 4x4 v8f accumulators, K stepped by 4.
// Per k-step: 4 A-frag loads (b64) + 8 B-frag loads feed 16 WMMAs (B reused
// across the 4 row tiles, A reused across the 4 col tiles).
// A 16x4 frag: lane<16 holds A[m][k],A[m][k+1]; lane>=16 holds A[m][k+2],A[m][k+3].
// B 4x16 frag: VGPR j, lanes0-15 = B[k+j][n], lanes16-31 = B[k+j+2][n].
// C/D: VGPR v, lanes0-15 -> (M=v, N=lane), lanes16-31 -> (M=v+8, N=lane-16).
// ---------------------------------------------------------------------------
__global__ __launch_bounds__(128)
void gemm_proj_wmma(const float* __restrict__ feats, const float* __restrict__ W,
                    float* __restrict__ ft, int N) {
  const int lane  = threadIdx.x & 31;
  const int wave  = threadIdx.x >> 5;          // 0..3 -> 64-col group
  const int row0  = blockIdx.x * 64;
  const int col0  = wave * 64;
  const int mrow  = lane & 15;
  const int khalf = (lane >> 4) * 2;           // 0 or 2

  const float* arow[4];
#pragma unroll
  for (int m = 0; m < 4; ++m) {
    int r = row0 + m * 16 + mrow;
    if (r >= N) r = N - 1;                     // tail clamp (loads only)
    arow[m] = feats + (size_t)r * DIN;
  }

  v8f acc[4][4];
#pragma unroll
  for (int m = 0; m < 4; ++m)
#pragma unroll
    for (int n = 0; n < 4; ++n) acc[m][n] = (v8f){};

  for (int k = 0; k < DIN; k += 4) {
    v2f a[4];
#pragma unroll
    for (int m = 0; m < 4; ++m) {
      a[m].x = arow[m][k + khalf];
      a[m].y = arow[m][k + khalf + 1];
    }

    const float* b0 = W + (size_t)(k + khalf) * HD + col0 + mrow;  // row k+khalf
    const float* b1 = b0 + HD;                                     // row k+khalf+1
    v2f b[4];
#pragma unroll
    for (int n = 0; n < 4; ++n) { b[n].x = b0[n * 16]; b[n].y = b1[n * 16]; }

#pragma unroll
    for (int m = 0; m < 4; ++m)
#pragma unroll
      for (int n = 0; n < 4; ++n)
        acc[m][n] = __builtin_amdgcn_wmma_f32_16x16x4_f32(
            false, a[m], false, b[n], (short)0, acc[m][n], false, false);
  }

#pragma unroll
  for (int m = 0; m < 4; ++m) {
    const int rbase = row0 + m * 16 + ((lane >> 4) << 3);  // +0 or +8
    float* out = ft + (size_t)rbase * HD + col0 + mrow;
#pragma unroll
    for (int v = 0; v < 8; ++v) {
      if (rbase + v < N) {
#pragma unroll
        for (int n = 0; n < 4; ++n) out[(size_t)v * HD + n * 16] = acc[m][n][v];
      }
    }
  }
}

// ---------------------------------------------------------------------------
// out[n*H+h] = sum_d ft[n][h*32+d] * attn[h][d]
// ---------------------------------------------------------------------------
__global__ void attn_dot_kernel(const float* __restrict__ ft,
                                const float* __restrict__ attn,
                                float* __restrict__ out, int N) {
  int idx = blockIdx.x * blockDim.x + threadIdx.x;   // n*H + h
  if (idx >= N * HEADS) return;
  int h = idx & (HEADS - 1);
  const float4* f  = (const float4*)(ft + (size_t)idx * DIMD);  // idx*32 == n*256 + h*32
  const float4* av = (const float4*)(attn + h * DIMD);
  float s = 0.f;
#pragma unroll
  for (int i = 0; i < 8; ++i) {
    float4 a = f[i], b = av[i];
    s += a.x * b.x + a.y * b.y + a.z * b.z + a.w * b.w;
  }
  out[idx] = s;
}

// ---------------------------------------------------------------------------
// init: rst = 0, emax = -inf, esum = 0
// ---------------------------------------------------------------------------
__global__ void init_kernel(float* __restrict__ rst, float* __restrict__ emax,
                            float* __restrict__ esum, int Nm) {
  size_t stride = (size_t)gridDim.x * blockDim.x;
  size_t idx = (size_t)blockIdx.x * blockDim.x + threadIdx.x;
  size_t total = (size_t)Nm * HD;
  for (size_t i = idx; i < total; i += stride) rst[i] = 0.f;
  size_t nh = (size_t)Nm * HEADS;
  for (size_t i = idx; i < nh; i += stride) { emax[i] = -INFINITY; esum[i] = 0.f; }
}

// order-preserving float atomic max (lowers to int atomic max/min)
__device__ inline void atomicMaxFloat(float* addr, float val) {
  if (val >= 0.f)
    atomicMax((int*)addr, __float_as_int(val));
  else
    atomicMin((unsigned int*)addr, __float_as_uint(val));
}

// ---------------------------------------------------------------------------
// edge logits + segment max:  e = lrelu(el[src] + er[dst]); emax[dst] max= e
// ---------------------------------------------------------------------------
__global__ void edge_logits_kernel(const int* __restrict__ src, const int* __restrict__ dst,
                                   const float* __restrict__ el, const float* __restrict__ er,
                                   float* __restrict__ ebuf, float* __restrict__ emax, int E) {
  int idx = blockIdx.x * blockDim.x + threadIdx.x;  // e*H + h
  if (idx >= E * HEADS) return;
  int e = idx >> 3, h = idx & (HEADS - 1);
  int s = src[e], d = dst[e];
  float v = el[s * HEADS + h] + er[d * HEADS + h];
  v = (v >= 0.f) ? v : NEG_SLOPE * v;
  ebuf[idx] = v;
  atomicMaxFloat(&emax[d * HEADS + h], v);
}

// ---------------------------------------------------------------------------
// ee = exp(e - emax[dst]); esum[dst] += ee  (ebuf overwritten with ee)
// ---------------------------------------------------------------------------
__global__ void edge_exp_kernel(const int* __restrict__ dst, const float* __restrict__ emax,
                                float* __restrict__ ebuf, float* __restrict__ esum, int E) {
  int idx = blockIdx.x * blockDim.x + threadIdx.x;
  if (idx >= E * HEADS) return;
  int e = idx >> 3, h = idx & (HEADS - 1);
  int d = dst[e];
  float ee = __expf(ebuf[idx] - emax[d * HEADS + h]);
  ebuf[idx] = ee;
  atomicAdd(&esum[d * HEADS + h], ee);
}

// ---------------------------------------------------------------------------
// scatter: one wave per edge (8 edges / 256-thread block). Lane l handles the
// 8 elements {h*32+l}; per head the wave-uniform weight w = ee/esum[dst][h].
// Fully coalesced gather of ft_src and global_atomic_add_f32 scatter into rst.
// ---------------------------------------------------------------------------
__global__ __launch_bounds__(256)
void edge_scatter_kernel(const int* __restrict__ src, const int* __restrict__ dst,
                         const float* __restrict__ ebuf, const float* __restrict__ esum,
                         const float* __restrict__ ft, float* __restrict__ rst, int E) {
  int e = blockIdx.x * 8 + (threadIdx.x >> 5);
  if (e >= E) return;
  int lane = threadIdx.x & 31;
  int s = src[e], d = dst[e];
  const float* fsrc = ft + (size_t)s * HD;
  float* out = rst + (size_t)d * HD;
#pragma unroll
  for (int h = 0; h < HEADS; ++h) {
    float w = ebuf[(size_t)e * HEADS + h] / esum[d * HEADS + h];
    atomicAdd(&out[h * 32 + lane], fsrc[h * 32 + lane] * w);
  }
}

// ---------------------------------------------------------------------------
extern "C" void kernel_launch(void* const* d_in, const int* in_sizes, int n_in,
                              void* d_out, int out_size, void* d_ws, size_t ws_size,
                              hipStream_t stream) {
  const float* master = (const float*)d_in[0];
  const float* af0    = (const float*)d_in[1];
  const float* af1    = (const float*)d_in[2];
  const int*   src0   = (const int*)d_in[3];
  const int*   dst0   = (const int*)d_in[4];
  const int*   src1   = (const int*)d_in[5];
  const int*   dst1   = (const int*)d_in[6];
  const float* Wm     = (const float*)d_in[7];
  const float* W0     = (const float*)d_in[8];
  const float* W1     = (const float*)d_in[9];
  const float* attn_l = (const float*)d_in[10];
  const float* attn_r = (const float*)d_in[11];

  const int Nm = in_sizes[0] / DIN;
  const int Na = in_sizes[1] / DIN;
  const int E0 = in_sizes[3];
  const int E1 = in_sizes[5];
  float* rst = (float*)d_out;

  // workspace carve-up (floats)
  float* ws   = (float*)d_ws;
  float* ft_m = ws;  ws += (size_t)Nm * HD;
  float* ft0  = ws;  ws += (size_t)Na * HD;
  float* ft1  = ws;  ws += (size_t)Na * HD;
  float* er   = ws;  ws += (size_t)Nm * HEADS;
  float* el0  = ws;  ws += (size_t)Na * HEADS;
  float* el1  = ws;  ws += (size_t)Na * HEADS;
  float* emax = ws;  ws += (size_t)Nm * HEADS;
  float* esum = ws;  ws += (size_t)Nm * HEADS;
  float* ebuf = ws;  // (E0+E1)*HEADS

  // 1) projections via fp32 WMMA (64x64 tile per wave)
  gemm_proj_wmma<<<(Nm + 63) / 64, 128, 0, stream>>>(master, Wm, ft_m, Nm);
  gemm_proj_wmma<<<(Na + 63) / 64, 128, 0, stream>>>(af0, W0, ft0, Na);
  gemm_proj_wmma<<<(Na + 63) / 64, 128, 0, stream>>>(af1, W1, ft1, Na);

  // 2) attention logits per node
  attn_dot_kernel<<<(Nm * HEADS + 255) / 256, 256, 0, stream>>>(ft_m, attn_r, er, Nm);
  attn_dot_kernel<<<(Na * HEADS + 255) / 256, 256, 0, stream>>>(ft0, attn_l, el0, Na);
  attn_dot_kernel<<<(Na * HEADS + 255) / 256, 256, 0, stream>>>(ft1, attn_l, el1, Na);

  // 3) init rst / emax / esum
  init_kernel<<<2048, 256, 0, stream>>>(rst, emax, esum, Nm);

  // 4) edge logits + segment max (both edge types into shared emax)
  edge_logits_kernel<<<(E0 * HEADS + 255) / 256, 256, 0, stream>>>(src0, dst0, el0, er, ebuf, emax, E0);
  edge_logits_kernel<<<(E1 * HEADS + 255) / 256, 256, 0, stream>>>(src1, dst1, el1, er,
                                                                   ebuf + (size_t)E0 * HEADS, emax, E1);

  // 5) exp + segment sum
  edge_exp_kernel<<<(E0 * HEADS + 255) / 256, 256, 0, stream>>>(dst0, emax, ebuf, esum, E0);
  edge_exp_kernel<<<(E1 * HEADS + 255) / 256, 256, 0, stream>>>(dst1, emax,
                                                                ebuf + (size_t)E0 * HEADS, esum, E1);

  // 6) normalized weighted scatter-add onto master nodes (1 wave per edge)
  edge_scatter_kernel<<<(E0 + 7) / 8, 256, 0, stream>>>(src0, dst0, ebuf, esum, ft0, rst, E0);
  edge_scatter_kernel<<<(E1 + 7) / 8, 256, 0, stream>>>(src1, dst1, ebuf + (size_t)E0 * HEADS,
                                                        esum, ft1, rst, E1);
}